// MLP_Weight_Pool_33741263078237
// MI455X (gfx1250) — compile-verified
//
#include <hip/hip_runtime.h>

#define N_TOKENS 32768
#define IN_DIM   128
#define OUT_DIM  128
#define N_NODES  256

typedef __attribute__((ext_vector_type(2))) float v2f;
typedef __attribute__((ext_vector_type(8))) float v8f;

// ---------------- bucketing kernels ----------------

__global__ __launch_bounds__(N_NODES) void zero_counts_kernel(int* counts) {
    counts[threadIdx.x] = 0;
}

__global__ __launch_bounds__(256) void count_kernel(const int* __restrict__ idx,
                                                    int* __restrict__ counts) {
    int t = blockIdx.x * 256 + threadIdx.x;
    if (t < N_TOKENS) atomicAdd(&counts[idx[t]], 1);
}

__global__ __launch_bounds__(N_NODES) void scan_kernel(const int* __restrict__ counts,
                                                       int* __restrict__ offsets,
                                                       int* __restrict__ cursor) {
    __shared__ int tmp[N_NODES];
    int i = threadIdx.x;
    int c = counts[i];
    tmp[i] = c;
    __syncthreads();
    for (int off = 1; off < N_NODES; off <<= 1) {
        int v = (i >= off) ? tmp[i - off] : 0;
        __syncthreads();
        tmp[i] += v;
        __syncthreads();
    }
    int excl = tmp[i] - c;     // exclusive prefix sum
    offsets[i] = excl;
    cursor[i]  = excl;
}

__global__ __launch_bounds__(256) void scatter_kernel(const int* __restrict__ idx,
                                                      int* __restrict__ cursor,
                                                      int* __restrict__ perm) {
    int t = blockIdx.x * 256 + threadIdx.x;
    if (t < N_TOKENS) {
        int node = idx[t];
        int pos = atomicAdd(&cursor[node], 1);
        perm[pos] = t;
    }
}

// ---------------- per-node GEMM with fp32 WMMA ----------------
//
// One block per node. W[node] (128x128 f32 = 64KB) staged in LDS as
// float2 pairs along K: wl[p*128 + col] = { W[2p][col], W[2p+1][col] }.
// Each wave computes 16-token x 128-out tiles with V_WMMA_F32_16X16X4_F32.
//
// Operand layouts (ISA 7.12.2, 32-bit, wave32):
//   A 16x4:  VGPR0 = K{0}(lanes0-15)/K{2}(lanes16-31), VGPR1 = K{1}/K{3}; M = lane&15
//   B 4x16:  mirrored: VGPR0 = row K{0}/K{2}, VGPR1 = row K{1}/K{3}; N = lane&15
//   C/D:     VGPR j = row M=j (lanes0-15) / M=j+8 (lanes16-31); N = lane&15

__global__ __launch_bounds__(256) void gemm_kernel(
    const float* __restrict__ x,
    const float* __restrict__ wpool,
    const float* __restrict__ bpool,
    const int*   __restrict__ perm,
    const int*   __restrict__ offsets,
    const int*   __restrict__ counts,
    float*       __restrict__ out)
{
    __shared__ float2 wl[(IN_DIM / 2) * OUT_DIM];   // 64 KB

    const int node = blockIdx.x;
    const float* wg = wpool + (size_t)node * IN_DIM * OUT_DIM;
    const int tid = threadIdx.x;

    // cooperative load of W[node] into LDS, pre-paired along K
    for (int i = tid; i < (IN_DIM / 2) * OUT_DIM; i += 256) {
        int p   = i >> 7;     // k-pair index
        int col = i & 127;    // output column
        wl[i] = make_float2(wg[(2 * p) * OUT_DIM + col],
                            wg[(2 * p + 1) * OUT_DIM + col]);
    }
    __syncthreads();

    const int start = offsets[node];
    const int cnt   = counts[node];
    const int lane  = tid & 31;
    const int wave  = tid >> 5;
    const int half  = lane >> 4;   // which 16-lane half
    const int m     = lane & 15;   // row (A) / column (B,D) index within half

    // per-lane bias registers for columns nt*16 + m
    float biasr[8];
#pragma unroll
    for (int nt = 0; nt < 8; ++nt)
        biasr[nt] = bpool[node * OUT_DIM + nt * 16 + m];

    // each of the 8 waves owns tiles wave, wave+8, ... (16 tokens per tile)
    for (int tile = wave; tile * 16 < cnt; tile += 8) {
        const int rbase = tile * 16;

        // A row for this lane: token at row (rbase + m); clamp ragged tail to a
        // valid token — its garbage only lands in D rows we never store.
        int rm = rbase + m;
        if (rm >= cnt) rm = cnt - 1;
        const int tokm = perm[start + rm];
        const float2* xr2 = (const float2*)(x + (size_t)tokm * IN_DIM);

        v8f acc[8] = {};   // 16x128 output tile, 8 column-tiles of 16

#pragma unroll 4
        for (int k = 0; k < IN_DIM; k += 4) {
            const int kk = k + 2 * half;          // even
            const float2 av = xr2[kk >> 1];       // A: X[m][kk], X[m][kk+1]
            v2f a; a.x = av.x; a.y = av.y;
#pragma unroll
            for (int nt = 0; nt < 8; ++nt) {
                const float2 bv = wl[(kk >> 1) * OUT_DIM + nt * 16 + m]; // B: W[kk][col], W[kk+1][col]
                v2f b; b.x = bv.x; b.y = bv.y;
                acc[nt] = __builtin_amdgcn_wmma_f32_16x16x4_f32(
                    /*neg_a=*/false, a, /*neg_b=*/false, b,
                    /*c_mod=*/(short)0, acc[nt],
                    /*reuse_a=*/false, /*reuse_b=*/false);
            }
        }

        // store: lane holds D rows {j + 8*half}, columns {nt*16 + m}
#pragma unroll
        for (int j = 0; j < 8; ++j) {
            const int r = rbase + j + 8 * half;
            if (r < cnt) {
                const int tok = perm[start + r];
                float* orow = out + (size_t)tok * OUT_DIM;
#pragma unroll
                for (int nt = 0; nt < 8; ++nt)
                    orow[nt * 16 + m] = acc[nt][j] + biasr[nt];
            }
        }
    }
}

// ---------------- launcher ----------------

extern "C" void kernel_launch(void* const* d_in, const int* in_sizes, int n_in,
                              void* d_out, int out_size, void* d_ws, size_t ws_size,
                              hipStream_t stream) {
    const float* x        = (const float*)d_in[0];
    const int*   node_ind = (const int*)d_in[1];   // [2, N_TOKENS] int32
    const float* wpool    = (const float*)d_in[2]; // [N_NODES, IN, OUT]
    const float* bpool    = (const float*)d_in[3]; // [N_NODES, OUT]
    float*       out      = (float*)d_out;

    const int* idx = node_ind + N_TOKENS;          // row 1 (reference uses node_ind[1])

    int* ws      = (int*)d_ws;
    int* counts  = ws;
    int* offsets = ws + N_NODES;
    int* cursor  = ws + 2 * N_NODES;
    int* perm    = ws + 3 * N_NODES;               // N_TOKENS ints

    zero_counts_kernel<<<1, N_NODES, 0, stream>>>(counts);
    count_kernel<<<N_TOKENS / 256, 256, 0, stream>>>(idx, counts);
    scan_kernel<<<1, N_NODES, 0, stream>>>(counts, offsets, cursor);
    scatter_kernel<<<N_TOKENS / 256, 256, 0, stream>>>(idx, cursor, perm);
    gemm_kernel<<<N_NODES, 256, 0, stream>>>(x, wpool, bpool, perm, offsets, counts, out);
}